// EnhancedGraphEncoder_78658031059101
// MI455X (gfx1250) — compile-verified
//
#include <hip/hip_runtime.h>

typedef float v2f __attribute__((ext_vector_type(2)));
typedef float v8f __attribute__((ext_vector_type(8)));

#define N_NODES   100000
#define N_EDGES   1600000
#define NUM_GRAPHS 128
#define FDIM      128
#define EPS_BN    1e-5f
#define MTILES    (N_NODES / 16)   // 6250, exact

// ---------------- utility ----------------
__global__ void k_zero(float* __restrict__ p, long long n) {
  long long t = (long long)blockIdx.x * blockDim.x + threadIdx.x;
  if (t < n) p[t] = 0.0f;
}

// ---------------- layer-1 edge aggregation (F=8) ----------------
// thread = (edge, feature f in 0..7); consecutive threads hit consecutive dst features
__global__ void k_edge_agg8(const float* __restrict__ x, const int* __restrict__ ei,
                            float* __restrict__ agg, float* __restrict__ deg) {
  long long t = (long long)blockIdx.x * blockDim.x + threadIdx.x;
  if (t >= (long long)N_EDGES * 8) return;
  long long e = t >> 3;
  int f = (int)(t & 7);
  int src = ei[e];
  int dst = ei[N_EDGES + e];
  atomicAdd(&agg[(long long)dst * 8 + f], x[(long long)src * 8 + f]);
  if (f == 0) atomicAdd(&deg[dst], 1.0f);
}

// ---------------- layer-1 SAGE linear (K=8, tiny) ----------------
// block = 128 threads = one node; x/agg rows staged through LDS
__global__ void k_sage1(const float* __restrict__ x, const float* __restrict__ agg,
                        const float* __restrict__ deg,
                        const float* __restrict__ Wl, const float* __restrict__ b,
                        const float* __restrict__ Wr, float* __restrict__ out) {
  __shared__ float xs[8], ms[8];
  int i = blockIdx.x;
  int j = threadIdx.x;
  if (j < 8) { xs[j] = x[(long long)i * 8 + j]; ms[j] = agg[(long long)i * 8 + j]; }
  __syncthreads();
  float inv = 1.0f / fmaxf(deg[i], 1.0f);
  float acc = b[j];
#pragma unroll
  for (int k = 0; k < 8; ++k)
    acc += ms[k] * inv * Wl[j * 8 + k] + xs[k] * Wr[j * 8 + k];
  out[(long long)i * FDIM + j] = acc;
}

// ---------------- BN stats: per-feature sum & sum-of-squares ----------------
__global__ void k_bn_stats(const float* __restrict__ h, float* __restrict__ st) {
  int j = threadIdx.x;  // 128 threads
  float s = 0.0f, s2 = 0.0f;
  for (long long i = blockIdx.x; i < N_NODES; i += gridDim.x) {
    float v = h[i * FDIM + j];
    s += v;
    s2 += v * v;
  }
  atomicAdd(&st[j], s);
  atomicAdd(&st[FDIM + j], s2);
}

// ---------------- BN apply + ReLU (in place allowed) ----------------
__global__ void k_bn_relu(const float* __restrict__ hpre, const float* __restrict__ st,
                          const float* __restrict__ g, const float* __restrict__ beta,
                          float* __restrict__ out) {
  long long t = (long long)blockIdx.x * blockDim.x + threadIdx.x;
  if (t >= (long long)N_NODES * FDIM) return;
  int j = (int)(t & (FDIM - 1));
  const float invN = 1.0f / (float)N_NODES;
  float mu = st[j] * invN;
  float var = st[FDIM + j] * invN - mu * mu;
  float y = g[j] * (hpre[t] - mu) * rsqrtf(var + EPS_BN) + beta[j];
  out[t] = fmaxf(y, 0.0f);
}

// ---------------- BN apply + ReLU + residual add (h = h1 + relu(bn(h2pre))) ----------------
__global__ void k_bn_relu_add(const float* __restrict__ hpre, const float* __restrict__ st,
                              const float* __restrict__ g, const float* __restrict__ beta,
                              const float* __restrict__ addin, float* __restrict__ out) {
  long long t = (long long)blockIdx.x * blockDim.x + threadIdx.x;
  if (t >= (long long)N_NODES * FDIM) return;
  int j = (int)(t & (FDIM - 1));
  const float invN = 1.0f / (float)N_NODES;
  float mu = st[j] * invN;
  float var = st[FDIM + j] * invN - mu * mu;
  float y = g[j] * (hpre[t] - mu) * rsqrtf(var + EPS_BN) + beta[j];
  out[t] = addin[t] + fmaxf(y, 0.0f);
}

// ---------------- layer-2 edge aggregation (F=128) ----------------
__global__ void k_edge_agg128(const float* __restrict__ h1, const int* __restrict__ ei,
                              float* __restrict__ agg) {
  long long t = (long long)blockIdx.x * blockDim.x + threadIdx.x;
  if (t >= (long long)N_EDGES * FDIM) return;
  long long e = t >> 7;
  int f = (int)(t & (FDIM - 1));
  int src = ei[e];
  int dst = ei[N_EDGES + e];
  atomicAdd(&agg[(long long)dst * FDIM + f], h1[(long long)src * FDIM + f]);
}

// ---------------- mean-normalize agg2 by degree (in place) ----------------
__global__ void k_normalize(float* __restrict__ agg, const float* __restrict__ deg) {
  long long t = (long long)blockIdx.x * blockDim.x + threadIdx.x;
  if (t >= (long long)N_NODES * FDIM) return;
  long long i = t >> 7;
  agg[t] *= 1.0f / fmaxf(deg[i], 1.0f);
}

// ---------------- layer-2 dual GEMM via V_WMMA_F32_16X16X4_F32 ----------------
// out[m][n] = sum_k A[m][k]*Wl[n][k] + sum_k B[m][k]*Wr[n][k] + bias[n]
// One wave computes a full 16x128 row block (8 N-tiles), A loads reused across N-tiles.
__global__ void __launch_bounds__(256)
k_gemm2_wmma(const float* __restrict__ A, const float* __restrict__ B,
             const float* __restrict__ Wl, const float* __restrict__ Wr,
             const float* __restrict__ bias, float* __restrict__ out) {
  int wave = threadIdx.x >> 5;
  int mtile = blockIdx.x * 8 + wave;
  if (mtile >= MTILES) return;            // wave-uniform guard: EXEC stays all-ones
  int lane = threadIdx.x & 31;
  int half = lane >> 4;                   // 0: K={0,1} rows, 1: K={2,3} rows
  int l = lane & 15;
  long long m0 = (long long)mtile * 16;

  // per-lane A/B base: row (m0+l), columns k + 2*half (float2, 8B aligned)
  const float* arow = A + (m0 + l) * FDIM + 2 * half;
  const float* brow = B + (m0 + l) * FDIM + 2 * half;

  v8f zero = {};
  v8f acc[8];
#pragma unroll
  for (int nt = 0; nt < 8; ++nt) acc[nt] = zero;

  for (int k = 0; k < FDIM; k += 4) {
    v2f aL = *(const v2f*)(arow + k);     // A tile 16x4 (f32), striped per ISA layout
    v2f aR = *(const v2f*)(brow + k);
#pragma unroll
    for (int nt = 0; nt < 8; ++nt) {
      int n = nt * 16 + l;
      v2f wl = *(const v2f*)(Wl + (long long)n * FDIM + k + 2 * half);  // B[k][n]=Wl[n][k]
      v2f wr = *(const v2f*)(Wr + (long long)n * FDIM + k + 2 * half);
      acc[nt] = __builtin_amdgcn_wmma_f32_16x16x4_f32(
          false, aL, false, wl, (short)0, acc[nt], false, false);
      acc[nt] = __builtin_amdgcn_wmma_f32_16x16x4_f32(
          false, aR, false, wr, (short)0, acc[nt], false, false);
    }
  }

#pragma unroll
  for (int nt = 0; nt < 8; ++nt) {
    int n = nt * 16 + l;
    float bb = bias[n];
#pragma unroll
    for (int r = 0; r < 8; ++r) {
      long long row = m0 + r + 8 * half;  // D layout: VGPR r -> M = r + 8*half
      out[row * FDIM + n] = acc[nt][r] + bb;
    }
  }
}

// ---------------- global mean pool (segment sum via atomics) ----------------
__global__ void k_pool(const float* __restrict__ h, const int* __restrict__ batch,
                       float* __restrict__ pool, float* __restrict__ cnt) {
  long long t = (long long)blockIdx.x * blockDim.x + threadIdx.x;
  if (t >= (long long)N_NODES * FDIM) return;
  long long i = t >> 7;
  int j = (int)(t & (FDIM - 1));
  int g = batch[i];
  atomicAdd(&pool[(long long)g * FDIM + j], h[t]);
  if (j == 0) atomicAdd(&cnt[g], 1.0f);
}

__global__ void k_finalize(const float* __restrict__ pool, const float* __restrict__ cnt,
                           float* __restrict__ out) {
  int t = blockIdx.x * blockDim.x + threadIdx.x;
  if (t >= NUM_GRAPHS * FDIM) return;
  int g = t >> 7;
  out[t] = pool[t] / fmaxf(cnt[g], 1.0f);
}

// ---------------- host orchestration ----------------
extern "C" void kernel_launch(void* const* d_in, const int* in_sizes, int n_in,
                              void* d_out, int out_size, void* d_ws, size_t ws_size,
                              hipStream_t stream) {
  const float* x     = (const float*)d_in[0];
  const int*   ei    = (const int*)d_in[1];    // [2, E]: row0=src, row1=dst
  const int*   batch = (const int*)d_in[2];
  const float* W1l   = (const float*)d_in[3];
  const float* b1    = (const float*)d_in[4];
  const float* W1r   = (const float*)d_in[5];
  const float* g1    = (const float*)d_in[6];
  const float* be1   = (const float*)d_in[7];
  const float* W2l   = (const float*)d_in[8];
  const float* b2    = (const float*)d_in[9];
  const float* W2r   = (const float*)d_in[10];
  const float* g2    = (const float*)d_in[11];
  const float* be2   = (const float*)d_in[12];
  float* out = (float*)d_out;

  float* ws = (float*)d_ws;
  const long long N = N_NODES, E = N_EDGES, G = NUM_GRAPHS, F = FDIM;
  float* agg1 = ws;                 // N*8
  float* deg  = agg1 + N * 8;       // N
  float* h1   = deg + N;            // N*128
  float* agg2 = h1 + N * F;         // N*128
  float* h2   = agg2 + N * F;       // N*128
  float* st   = h2 + N * F;         // 256 (sum, sumsq)
  float* pool = st + 2 * F;         // G*128
  float* cnt  = pool + G * F;       // G

  const int B256 = 256;
  const long long NF = N * F;       // 12,800,000
  const int gNF = (int)((NF + B256 - 1) / B256);

  // zero the accumulator regions (d_ws is poisoned)
  {
    long long n1 = N * 8 + N;       // agg1 + deg (contiguous)
    k_zero<<<(int)((n1 + B256 - 1) / B256), B256, 0, stream>>>(agg1, n1);
    k_zero<<<gNF, B256, 0, stream>>>(agg2, NF);
    k_zero<<<1, B256, 0, stream>>>(st, 2 * F);
    long long n2 = G * F + G;       // pool + cnt (contiguous)
    k_zero<<<(int)((n2 + B256 - 1) / B256), B256, 0, stream>>>(pool, n2);
  }

  // ---- layer 1 ----
  {
    long long t = E * 8;
    k_edge_agg8<<<(int)((t + B256 - 1) / B256), B256, 0, stream>>>(x, ei, agg1, deg);
  }
  k_sage1<<<(int)N, FDIM, 0, stream>>>(x, agg1, deg, W1l, b1, W1r, h1);  // h1 := h1pre
  k_bn_stats<<<512, FDIM, 0, stream>>>(h1, st);
  k_bn_relu<<<gNF, B256, 0, stream>>>(h1, st, g1, be1, h1);              // h1 := relu(bn)

  // ---- layer 2 ----
  {
    long long t = E * F;
    k_edge_agg128<<<(int)((t + B256 - 1) / B256), B256, 0, stream>>>(h1, ei, agg2);
  }
  k_normalize<<<gNF, B256, 0, stream>>>(agg2, deg);
  k_gemm2_wmma<<<(MTILES + 7) / 8, 256, 0, stream>>>(agg2, h1, W2l, W2r, b2, h2);
  k_zero<<<1, B256, 0, stream>>>(st, 2 * F);
  k_bn_stats<<<512, FDIM, 0, stream>>>(h2, st);
  k_bn_relu_add<<<gNF, B256, 0, stream>>>(h2, st, g2, be2, h1, h2);      // h2 := h1 + relu(bn(h2))

  // ---- pooling ----
  k_pool<<<gNF, B256, 0, stream>>>(h2, batch, pool, cnt);
  k_finalize<<<(NUM_GRAPHS * FDIM + B256 - 1) / B256, B256, 0, stream>>>(pool, cnt, out);
}